// Attention_14688788152633
// MI455X (gfx1250) — compile-verified
//
#include <hip/hip_runtime.h>

// MI455X / gfx1250, wave32. fp32 WMMA path: V_WMMA_F32_16X16X4_F32.
typedef __attribute__((ext_vector_type(2))) float f32x2;
typedef __attribute__((ext_vector_type(8))) float f32x8;

#define N_NODES   16384
#define B_EDGES   500000
#define F_DIM     128
#define TILE_M    16
#define NTILES    (B_EDGES / TILE_M)          // 31250, exact
#define WAVES_PB  8
#define THREADS   (WAVES_PB * 32)             // 256
#define KT_PITCH  130                          // pad to kill LDS bank conflicts
#define TP_PITCH  130
#define KT_FLOATS (F_DIM * KT_PITCH)          // 16640
#define TP_FLOATS (TILE_M * TP_PITCH)         // 2080
#define LDS_BYTES ((KT_FLOATS + WAVES_PB * TP_FLOATS) * 4)  // 133,120 B < 320 KB/WGP

__global__ __launch_bounds__(THREADS)
void attention_edge_wmma_kernel(const float* __restrict__ t_pos,
                                const float* __restrict__ t_neg,
                                const float* __restrict__ A_in,
                                const float* __restrict__ kernelW,
                                const float* __restrict__ bias,
                                const int*   __restrict__ h_idx,
                                const int*   __restrict__ t_idx,
                                float* __restrict__ out_h,
                                float* __restrict__ out_pos,
                                float* __restrict__ out_neg)
{
    extern __shared__ float smem[];
    float* kT = smem;                               // kernel^T, [128][130]

    const int tid  = threadIdx.x;
    const int lane = tid & 31;
    const int wave = tid >> 5;
    const int q    = lane & 15;                     // column-in-tile / row-in-tile selector
    const int hi   = lane >> 4;                     // half-wave id

    // ---- cooperative stage of kernel^T into LDS (all 256 threads) ----
    // kernel is row-major [K=128][N=128]; store transposed so B-fragments
    // (two consecutive K for fixed N) are a single 8B-aligned ds_load_b64.
    #pragma unroll 8
    for (int i = 0; i < (F_DIM * F_DIM) / THREADS; ++i) {
        int idx = i * THREADS + tid;
        int k = idx >> 7;
        int n = idx & 127;
        kT[n * KT_PITCH + k] = kernelW[idx];
    }
    __syncthreads();

    const int tile = blockIdx.x * WAVES_PB + wave;  // wave-uniform
    if (tile >= NTILES) return;                     // after the only barrier
    const int r0 = tile * TILE_M;

    float* tp = smem + KT_FLOATS + wave * TP_FLOATS; // this wave's 16x128 tile

    // ---- stage 16x128 t_pos tile into LDS (coalesced float2 loads) ----
    const f32x2* tpg = (const f32x2*)(t_pos + (size_t)r0 * F_DIM);
    #pragma unroll 8
    for (int i = 0; i < 32; ++i) {
        int idx2 = i * 32 + lane;                   // 0..1023 float2s
        int row  = idx2 >> 6;                       // 64 float2 per row
        int c2   = idx2 & 63;
        f32x2 v  = tpg[idx2];
        *(f32x2*)&tp[row * TP_PITCH + c2 * 2] = v;  // same-wave LDS ops are in-order
    }

    // per-lane bias for this lane's 8 output columns (c = 16n + q)
    float biasl[8];
    #pragma unroll
    for (int n = 0; n < 8; ++n) biasl[n] = bias[n * 16 + q];

    // per-edge attention gather: lane l holds att for tile row (l & 15)
    float att_l;
    {
        int gr = r0 + q;
        int hrow = h_idx[gr];
        int tcol = t_idx[gr];
        att_l = A_in[(size_t)hrow * N_NODES + tcol];
    }

    // ---- GEMM: t_V[16x128] = tile[16x128] @ kernel[128x128] via WMMA f32 ----
    f32x8 acc[8];
    #pragma unroll
    for (int n = 0; n < 8; ++n)
        #pragma unroll
        for (int j = 0; j < 8; ++j) acc[n][j] = 0.0f;

    #pragma unroll 4
    for (int k = 0; k < 32; ++k) {
        // A 16x4 fragment: lane = row q, K-pair base 2*hi
        f32x2 a = *(const f32x2*)&tp[q * TP_PITCH + k * 4 + hi * 2];
        #pragma unroll
        for (int n = 0; n < 8; ++n) {
            // B 4x16 fragment: N = 16n + q, K = 4k + 2*hi + {0,1}
            f32x2 b = *(const f32x2*)&kT[(n * 16 + q) * KT_PITCH + k * 4 + hi * 2];
            acc[n] = __builtin_amdgcn_wmma_f32_16x16x4_f32(
                false, a, false, b, (short)0, acc[n], false, false);
        }
    }

    // ---- epilogue: h = relu(att * t_V + bias); logits = <h, t_pos/neg> ----
    // C/D layout: VGPR r, lane l -> row m = r + 8*(l/16), col = 16n + (l%16)
    float pos_acc[8], neg_acc[8];
    #pragma unroll
    for (int r = 0; r < 8; ++r) {
        int m = r + hi * 8;
        // broadcast att of row m from the lane that gathered it (same half)
        float att_r = __shfl(att_l, (lane & 16) + m, 32);
        size_t gr = (size_t)(r0 + m);
        const float* tneg_row = t_neg + gr * F_DIM;
        float*       h_row    = out_h + gr * F_DIM;
        float pa = 0.0f, na = 0.0f;
        #pragma unroll
        for (int n = 0; n < 8; ++n) {
            int c = n * 16 + q;
            float hval = fmaxf(fmaf(acc[n][r], att_r, biasl[n]), 0.0f);
            h_row[c] = hval;                        // 64B-coalesced per half-wave
            pa = fmaf(hval, tp[m * TP_PITCH + c], pa);
            na = fmaf(hval, tneg_row[c], na);
        }
        pos_acc[r] = pa;
        neg_acc[r] = na;
    }

    // reduce the 8-column partials across each 16-lane group (masks stay in-half)
    #pragma unroll
    for (int r = 0; r < 8; ++r) {
        float p = pos_acc[r], g = neg_acc[r];
        #pragma unroll
        for (int mask = 1; mask <= 8; mask <<= 1) {
            p += __shfl_xor(p, mask, 32);
            g += __shfl_xor(g, mask, 32);
        }
        if (q == 0) {                               // lanes 0 and 16 write
            int row = r0 + r + hi * 8;
            out_pos[row] = p;
            out_neg[row] = g;
        }
    }
}

extern "C" void kernel_launch(void* const* d_in, const int* in_sizes, int n_in,
                              void* d_out, int out_size, void* d_ws, size_t ws_size,
                              hipStream_t stream) {
    // setup_inputs order: h_e(0, unused), t_pos_e(1), t_neg_e(2), A_in(3),
    //                     kernel(4), bias(5), h_indices(6), t_indices(7)
    const float* t_pos = (const float*)d_in[1];
    const float* t_neg = (const float*)d_in[2];
    const float* A_in  = (const float*)d_in[3];
    const float* kern  = (const float*)d_in[4];
    const float* bias  = (const float*)d_in[5];
    const int*   h_idx = (const int*)d_in[6];
    const int*   t_idx = (const int*)d_in[7];

    float* out_h   = (float*)d_out;                         // [B, 128]
    float* out_pos = out_h + (size_t)B_EDGES * F_DIM;       // [B]
    float* out_neg = out_pos + B_EDGES;                     // [B]

    int blocks = (NTILES + WAVES_PB - 1) / WAVES_PB;        // 3907
    attention_edge_wmma_kernel<<<blocks, THREADS, LDS_BYTES, stream>>>(
        t_pos, t_neg, A_in, kern, bias, h_idx, t_idx, out_h, out_pos, out_neg);
}